// BilinearInterpolation_45414984187978
// MI455X (gfx1250) — compile-verified
//
#include <hip/hip_runtime.h>

// Spatial-transformer bilinear sampling for MI455X (gfx1250).
//
// - Coordinate transform einsum('bij,jn->bin') done with V_WMMA_F32_16X16X4_F32:
//     A (16x4)  = 8 batches' theta rows (M = 2b+i), K columns 0..2 = theta, col 3 = 0
//     B (4x16)  = 16 pixel positions, column n = (xn, yn, 1, 0)
//     D (16x16) = row 2b+q gives x_t/y_t for batch b at pixel n  -> 128 pixels / WMMA
// - Gathers hit L2 (all 64 images = 113 MB < 192 MB L2); output is streamed with
//   non-temporal stores so the 113 MB of write traffic doesn't evict the images.

#define H_OUT 384
#define W_OUT 384
#define C_IMG 3
#define NPOS  (H_OUT * W_OUT)              // 147456
#define IMG_ELEMS ((size_t)NPOS * C_IMG)   // elems per image
#define BATCH 64

typedef __attribute__((ext_vector_type(2))) float v2f;
typedef __attribute__((ext_vector_type(8))) float v8f;

__global__ __launch_bounds__(256) void stn_bilinear_wmma(
    const float* __restrict__ images,   // [64, 384, 384, 3]
    const float* __restrict__ theta,    // [64, 2, 3]
    float* __restrict__ out)            // [64, 384, 384, 3]
{
    const int lane      = threadIdx.x & 31;
    const int waveInBlk = threadIdx.x >> 5;
    const int n   = lane & 15;    // matrix column == position slot
    const int hi  = lane >> 4;    // 0 = lanes 0-15, 1 = lanes 16-31
    const int bg  = blockIdx.y;   // batch group of 8
    const int pos = blockIdx.x * 128 + waveInBlk * 16 + n;

    // ---- A matrix (16x4): lanes 0-15 hold K=0,1 ; lanes 16-31 hold K=2,3
    // Row M = 2b + i holds theta[bg*8+b, i, :] (padded 4th column = 0).
    const int   bA = n >> 1;
    const int   iA = n & 1;
    const float* th = theta + ((size_t)(bg * 8 + bA)) * 6 + (size_t)iA * 3;
    v2f A;
    A.x = hi ? th[2] : th[0];
    A.y = hi ? 0.0f  : th[1];

    // ---- B matrix (4x16): column n = (xn, yn, 1, 0) for pixel 'pos'
    const int row = pos / W_OUT;
    const int col = pos - row * W_OUT;
    const float xn = -1.0f + (2.0f / (float)(W_OUT - 1)) * (float)col;
    const float yn = -1.0f + (2.0f / (float)(H_OUT - 1)) * (float)row;
    v2f B;
    B.x = hi ? 1.0f : xn;
    B.y = hi ? 0.0f : yn;

    v8f Cz = {0.f, 0.f, 0.f, 0.f, 0.f, 0.f, 0.f, 0.f};
    // D[M=2b+i, N=n] = sum_j theta[b,i,j] * coords[j,n]   (exact f32 matmul)
    v8f D = __builtin_amdgcn_wmma_f32_16x16x4_f32(
        /*neg_a=*/false, A, /*neg_b=*/false, B,
        /*c_mod=*/(short)0, Cz, /*reuse_a=*/false, /*reuse_b=*/false);

    // Lane n (lanes 0-15) holds D rows 0..7  -> batches bg*8 + 0..3
    // Lane 16+n           holds D rows 8..15 -> batches bg*8 + 4..7
    const int bBase = bg * 8 + hi * 4;

#pragma unroll
    for (int q = 0; q < 4; ++q) {
        const int   b  = bBase + q;
        const float xt = D[2 * q];
        const float yt = D[2 * q + 1];

        float x = (xt + 1.0f) * ((float)W_OUT * 0.5f);
        float y = (yt + 1.0f) * ((float)H_OUT * 0.5f);

        const float xf = floorf(x), yf = floorf(y);
        int x0 = (int)xf, y0 = (int)yf;
        int x1 = x0 + 1,  y1 = y0 + 1;
        x0 = x0 < 0 ? 0 : (x0 > W_OUT - 1 ? W_OUT - 1 : x0);
        x1 = x1 < 0 ? 0 : (x1 > W_OUT - 1 ? W_OUT - 1 : x1);
        y0 = y0 < 0 ? 0 : (y0 > H_OUT - 1 ? H_OUT - 1 : y0);
        y1 = y1 < 0 ? 0 : (y1 > H_OUT - 1 ? H_OUT - 1 : y1);
        const float xc = fminf(fmaxf(x, 0.0f), (float)(W_OUT - 1));
        const float yc = fminf(fmaxf(y, 0.0f), (float)(H_OUT - 1));

        const float x0f = (float)x0, x1f = (float)x1;
        const float y0f = (float)y0, y1f = (float)y1;
        const float wa = (x1f - xc) * (y1f - yc);
        const float wb = (x1f - xc) * (yc - y0f);
        const float wc = (xc - x0f) * (y1f - yc);
        const float wd = (xc - x0f) * (yc - y0f);

        const float* img = images + (size_t)b * IMG_ELEMS;
        const int ia = (y0 * W_OUT + x0) * C_IMG;
        const int ib = (y1 * W_OUT + x0) * C_IMG;
        const int ic = (y0 * W_OUT + x1) * C_IMG;
        const int id = (y1 * W_OUT + x1) * C_IMG;

        float* o = out + (size_t)b * IMG_ELEMS + (size_t)pos * C_IMG;
#pragma unroll
        for (int ch = 0; ch < 3; ++ch) {
            const float v = wa * img[ia + ch] + wb * img[ib + ch] +
                            wc * img[ic + ch] + wd * img[id + ch];
            __builtin_nontemporal_store(v, o + ch);  // stream output, keep images in L2
        }
    }
}

extern "C" void kernel_launch(void* const* d_in, const int* in_sizes, int n_in,
                              void* d_out, int out_size, void* d_ws, size_t ws_size,
                              hipStream_t stream) {
    const float* images = (const float*)d_in[0];  // [64,384,384,3] f32
    const float* theta  = (const float*)d_in[1];  // [64,2,3] f32
    float* out = (float*)d_out;                   // [64,384,384,3] f32

    // Each block: 8 waves x (16 positions x 8 batches) = 128 positions, 8 batches.
    dim3 grid(NPOS / 128, BATCH / 8);   // (1152, 8)
    dim3 block(256);
    stn_bilinear_wmma<<<grid, block, 0, stream>>>(images, theta, out);
}